// LaplaceAggregation_5772436045849
// MI455X (gfx1250) — compile-verified
//
#include <hip/hip_runtime.h>

#define T_LEN   1024
#define F_DIM   2048
#define FT      32          // f-channels per block
#define NCHUNK  8
#define CHUNK   (T_LEN / NCHUNK)   // 128

typedef __attribute__((ext_vector_type(4))) unsigned int v4u;
typedef __attribute__((ext_vector_type(8))) int          v8i;
typedef __attribute__((ext_vector_type(4))) int          v4i;
typedef __attribute__((ext_vector_type(2))) float        vf2;

// ---- Setup: lambda = exp(a+ib), Lambda = lambda^CHUNK, in fp64 -> fp32 tables ----
__launch_bounds__(256)
__global__ void laplace_setup_kernel(const double* __restrict__ a,
                                     const double* __restrict__ b,
                                     vf2* __restrict__ lam,      // [F_DIM]
                                     vf2* __restrict__ lamC)     // [F_DIM]
{
    const int f = blockIdx.x * blockDim.x + threadIdx.x;
    if (f >= F_DIM) return;
    const double ad = a[f];
    const double bd = b[f];
    const double ea = exp(ad);
    vf2 l;  l.x = (float)(ea * cos(bd));  l.y = (float)(ea * sin(bd));
    const double eC = exp((double)CHUNK * ad);
    const double tC = (double)CHUNK * bd;
    vf2 L;  L.x = (float)(eC * cos(tC));  L.y = (float)(eC * sin(tC));
    lam[f]  = l;
    lamC[f] = L;
}

__launch_bounds__(256)
__global__ void laplace_scan_kernel(const float* __restrict__ x,
                                    const float* __restrict__ mem_real,
                                    const float* __restrict__ mem_imag,
                                    const vf2* __restrict__ lam,
                                    const vf2* __restrict__ lamC,
                                    vf2* __restrict__ out)
{
    // Dynamic LDS: [0 .. 128KB) = x tile [T_LEN][FT], then chunk-end states.
    extern __shared__ float xs[];
    vf2* hend = (vf2*)(xs + (size_t)T_LEN * FT);   // [NCHUNK][FT]

    const int ftile = blockIdx.x;
    const int batch = blockIdx.y;
    const int f0 = ftile * FT;
    const int fl = threadIdx.x & (FT - 1);
    const int ck = threadIdx.x >> 5;               // wave id == time chunk
    const int f  = f0 + fl;

    // ---- TDM: DMA the 32(f) x 1024(t) fp32 tile of x into LDS (offset 0) ----
#if __has_builtin(__builtin_amdgcn_tensor_load_to_lds)
    if (threadIdx.x < 32) {   // wave 0 issues; TDM ignores EXEC, descriptor is wave-uniform
        unsigned long long gaddr =
            (unsigned long long)(const void*)(x + ((size_t)batch * T_LEN * F_DIM + f0));
        // D# group 0: count=1 | lds_addr=0 | global_addr | type=2
        v4u g0 = { 1u, 0u,
                   (unsigned)(gaddr & 0xFFFFFFFFu),
                   (unsigned)((gaddr >> 32) & 0x1FFFFFFu) | 0x80000000u };
        // D# group 1:
        //  d0: data_size=2 (4B)          -> 0x00020000
        //  d1: tensor_dim0[15:0]<<16     -> 2048 -> 0x08000000
        //  d2: tensor_dim1[15:0]<<16     -> 1024 -> 0x04000000
        //  d3: tile_dim0<<16             -> 32   -> 0x00200000
        //  d4: tile_dim1                 -> 1024 -> 0x00000400
        //  d5: tensor_dim0_stride[31:0]  -> 2048 -> 0x00000800
        v8i g1 = { 0x00020000, 0x08000000, 0x04000000, 0x00200000,
                   0x00000400, 0x00000800, 0, 0 };
        v4i g2 = { 0, 0, 0, 0 };   // 2D tensor: groups 2/3 unused
        v4i g3 = { 0, 0, 0, 0 };
        v8i g4 = { 0, 0, 0, 0, 0, 0, 0, 0 };   // 6-arg toolchain form: zero-filled
        __builtin_amdgcn_tensor_load_to_lds(g0, g1, g2, g3, g4, 0);
        __builtin_amdgcn_s_wait_tensorcnt(0);
    }
#else
    for (int t = ck; t < T_LEN; t += NCHUNK)
        xs[t * FT + fl] = x[(size_t)batch * T_LEN * F_DIM + (size_t)t * F_DIM + f];
#endif
    __syncthreads();

    // ---- per-channel decay factors from precomputed tables (coalesced b64 loads) ----
    const vf2 l = lam[f];
    const vf2 L = lamC[f];
    const float lr = l.x, li = l.y;
    const float Lr = L.x, Li = L.y;

    const float* xp = xs + (size_t)ck * CHUNK * FT + fl;

    // ---- Phase A: chunk-local scan from zero state -> chunk-end state ----
    float hr = 0.0f, hi = 0.0f;
#pragma unroll 8
    for (int t = 0; t < CHUNK; ++t) {
        const float xv = xp[t * FT];
        const float nr = fmaf(lr, hr, fmaf(-li, hi, xv));
        const float ni = fmaf(lr, hi, li * hr);
        hr = nr; hi = ni;
    }
    vf2 he; he.x = hr; he.y = hi;
    hend[ck * FT + fl] = he;
    __syncthreads();

    // ---- Carry: C = mem0; C = lambda^128 * C + hend[j] for preceding chunks ----
    const size_t mi = (size_t)batch * F_DIM + f;       // mem[B,1,F,1]
    float cr = mem_real[mi], ci = mem_imag[mi];
    for (int j = 0; j < ck; ++j) {
        const vf2 e = hend[j * FT + fl];
        const float nr = fmaf(Lr, cr, fmaf(-Li, ci, e.x));
        const float ni = fmaf(Lr, ci, fmaf(Li, cr, e.y));
        cr = nr; ci = ni;
    }

    // ---- Phase B: re-scan from LDS with correct init state; stream complex64 out (NT) ----
    hr = cr; hi = ci;
    vf2* op = out + (size_t)batch * T_LEN * F_DIM + (size_t)ck * CHUNK * F_DIM + f;
#pragma unroll 8
    for (int t = 0; t < CHUNK; ++t) {
        const float xv = xp[t * FT];
        const float nr = fmaf(lr, hr, fmaf(-li, hi, xv));
        const float ni = fmaf(lr, hi, li * hr);
        hr = nr; hi = ni;
        vf2 o; o.x = hr; o.y = hi;
        __builtin_nontemporal_store(o, op + (size_t)t * F_DIM);
    }
}

extern "C" void kernel_launch(void* const* d_in, const int* in_sizes, int n_in,
                              void* d_out, int out_size, void* d_ws, size_t ws_size,
                              hipStream_t stream) {
    const float*  x  = (const float*)d_in[0];
    const float*  mr = (const float*)d_in[1];
    const float*  mi = (const float*)d_in[2];
    const double* a  = (const double*)d_in[3];
    const double* b  = (const double*)d_in[4];
    vf2* out  = (vf2*)d_out;
    vf2* lam  = (vf2*)d_ws;            // F_DIM float2
    vf2* lamC = lam + F_DIM;           // F_DIM float2  (32 KB total in ws)

    laplace_setup_kernel<<<dim3((F_DIM + 255) / 256), dim3(256), 0, stream>>>(a, b, lam, lamC);

    const int B = in_sizes[0] / (T_LEN * F_DIM);   // = 8
    dim3 grid(F_DIM / FT, B);                      // 64 x 8 = 512 blocks
    dim3 block(FT * NCHUNK);                       // 256 threads = 8 waves
    const size_t smem = (size_t)T_LEN * FT * sizeof(float)      // 128 KB x tile
                      + (size_t)NCHUNK * FT * sizeof(vf2);      // 2 KB carries

    laplace_scan_kernel<<<grid, block, smem, stream>>>(x, mr, mi, lam, lamC, out);
}